// _RoIPooling2d_74363063763031
// MI455X (gfx1250) — compile-verified
//
#include <hip/hip_runtime.h>
#include <stdint.h>

// Problem constants (from the reference: features [4,256,38,38], rois [64,5])
#define PH_    7
#define PW_    7
#define C_     256
#define H_     38
#define W_     38
#define CB_    8            // channels staged per block via TDM
#define SCALE_ 0.0625f

typedef unsigned int u32x4 __attribute__((ext_vector_type(4)));
typedef int          i32x4 __attribute__((ext_vector_type(4)));
typedef int          i32x8 __attribute__((ext_vector_type(8)));

__device__ __forceinline__ int imax(int a, int b) { return a > b ? a : b; }
__device__ __forceinline__ int imin(int a, int b) { return a < b ? a : b; }

__device__ __forceinline__ void tdm_load_to_lds(u32x4 g0, i32x8 g1, i32x4 g2, i32x4 g3) {
#if __has_include(<hip/amd_detail/amd_gfx1250_TDM.h>)
  // amdgpu-toolchain (therock-10.0 headers): 6-arg builtin
  i32x8 g4 = {0, 0, 0, 0, 0, 0, 0, 0};
  __builtin_amdgcn_tensor_load_to_lds(g0, g1, g2, g3, g4, 0);
#else
  // ROCm 7.2 (clang-22): 5-arg builtin
  __builtin_amdgcn_tensor_load_to_lds(g0, g1, g2, g3, 0);
#endif
}

__global__ __launch_bounds__(256) void roipool_tdm_kernel(
    const float* __restrict__ feat,   // [B, C, H, W]
    const float* __restrict__ rois,   // [N, 5]  (b, x1, y1, x2, y2)
    float* __restrict__ out)          // [N, C, 7, 7]
{
  __shared__ float smem[CB_ * H_ * W_];   // 46,208 B: staged RoI tile, CB_ channels

  const int n   = blockIdx.x;
  const int c0  = blockIdx.y * CB_;
  const int tid = threadIdx.x;

  // ---- uniform per-RoI geometry (scalarizes to SALU: s_mul_f32/s_rndne_f32/s_cvt) ----
  const float* r = rois + n * 5;
  const int b  = (int)r[0];
  const int sw = (int)rintf(r[1] * SCALE_);   // rintf == round-half-even == jnp.round
  const int sh = (int)rintf(r[2] * SCALE_);
  const int ew = (int)rintf(r[3] * SCALE_);
  const int eh = (int)rintf(r[4] * SCALE_);
  const int rw = imax(ew - sw + 1, 1);
  const int rh = imax(eh - sh + 1, 1);

  // Staged rectangle = union of all bins: rows [clip(sh), clip(sh+rh)), cols likewise.
  const int row_lo = imin(imax(sh, 0), H_);
  const int row_hi = imin(imax(sh + rh, 0), H_);
  const int col_lo = imin(imax(sw, 0), W_);
  const int col_hi = imin(imax(sw + rw, 0), W_);
  const int rh_tot = row_hi - row_lo;
  const int rw_tot = col_hi - col_lo;
  const bool nonempty = (rh_tot > 0) && (rw_tot > 0);

  // ---- TDM: DMA the (rw_tot x rh_tot x CB_) strided tile into LDS ----
  // Wave-uniform guard: TDM ignores EXEC, so restrict issue to wave 0 via an SGPR
  // compare -> scalar branch (waves 1..7 statically skip; exactly one DMA per block).
  const int wave_id = __builtin_amdgcn_readfirstlane((int)threadIdx.x) >> 5;
  if (wave_id == 0 && nonempty) {
    const uint64_t gaddr = (uint64_t)(uintptr_t)(feat
        + ((size_t)b * C_ + (size_t)c0) * (size_t)(H_ * W_)
        + (size_t)row_lo * W_ + (size_t)col_lo);

    const uint32_t ga_lo = (uint32_t)__builtin_amdgcn_readfirstlane((int)(uint32_t)gaddr);
    const uint32_t ga_hi = (uint32_t)__builtin_amdgcn_readfirstlane((int)(uint32_t)(gaddr >> 32));
    const uint32_t t0    = (uint32_t)__builtin_amdgcn_readfirstlane(rw_tot);  // tile X (elems)
    const uint32_t t1    = (uint32_t)__builtin_amdgcn_readfirstlane(rh_tot);  // tile Y (rows)
    const uint32_t ldsb  = (uint32_t)(uintptr_t)&smem[0];  // low 32 bits = LDS byte offset

    // D# group 0: count=1 | lds_addr | global_addr[56:0] | type=2
    u32x4 g0;
    g0[0] = 1u;
    g0[1] = ldsb;
    g0[2] = ga_lo;
    g0[3] = (ga_hi & 0x01FFFFFFu) | (2u << 30);

    // D# group 1: data_size=4B; tensor dims huge (tile is pre-clipped, no OOB wanted);
    // tile dims (t0, t1, CB_); dim0 stride = W, dim1 stride = H*W (elements).
    const uint32_t BIG = 0x7FFFFFFFu;         // tensor_dim0/1/2 (OOB never triggers)
    const uint32_t s1  = (uint32_t)(H_ * W_); // tensor_dim1_stride (elements)
    i32x8 g1;
    g1[0] = (int)(2u << 16);                                          // data_size=4B, mask=0
    g1[1] = (int)((BIG & 0xFFFFu) << 16);                             // tensor_dim0[15:0]
    g1[2] = (int)(((BIG >> 16) & 0xFFFFu) | ((BIG & 0xFFFFu) << 16)); // dim0 hi | dim1 lo
    g1[3] = (int)(((BIG >> 16) & 0xFFFFu) | (t0 << 16));              // dim1 hi | tile_dim0
    g1[4] = (int)(t1 | ((uint32_t)CB_ << 16));                        // tile_dim1 | tile_dim2
    g1[5] = (int)W_;                                                  // stride0[31:0] = W
    g1[6] = (int)((s1 & 0xFFFFu) << 16);                              // stride0[47:32]=0 | stride1[15:0]
    g1[7] = (int)(s1 >> 16);                                          // stride1[47:16]

    // D# group 2: tensor_dim2 huge; iterate off; dim3 unused (tile_dim3 = 0)
    i32x4 g2; g2[0] = (int)BIG; g2[1] = 0; g2[2] = 0; g2[3] = 0;
    i32x4 g3 = {0, 0, 0, 0};

    tdm_load_to_lds(g0, g1, g2, g3);
    __builtin_amdgcn_s_wait_tensorcnt(0);   // data in LDS before we release the barrier
  }
  __syncthreads();

  // ---- compute 7x7 bin maxima for CB_ channels out of LDS ----
  const int out_base = (n * C_ + c0) * (PH_ * PW_);
  for (int idx = tid; idx < CB_ * PH_ * PW_; idx += 256) {
    const int ch  = idx / (PH_ * PW_);
    const int rem = idx - ch * (PH_ * PW_);
    const int ph  = rem / PW_;
    const int pw  = rem - ph * PW_;

    // floor/ceil bin bounds (integer math == reference f32 math for these magnitudes)
    const int hs = imin(imax((ph * rh) / PH_ + sh, 0), H_);
    const int he = imin(imax(((ph + 1) * rh + (PH_ - 1)) / PH_ + sh, 0), H_);
    const int ws = imin(imax((pw * rw) / PW_ + sw, 0), W_);
    const int we = imin(imax(((pw + 1) * rw + (PW_ - 1)) / PW_ + sw, 0), W_);

    const bool any = (hs < he) && (ws < we);
    float m = -__builtin_inff();
    if (any) {
      const float* base = smem + (size_t)ch * rh_tot * rw_tot;
      for (int h = hs; h < he; ++h) {
        const float* row = base + (size_t)(h - row_lo) * rw_tot - col_lo;
        for (int w = ws; w < we; ++w)
          m = fmaxf(m, row[w]);
      }
    }
    out[out_base + idx] = any ? m : 0.0f;   // empty bins -> 0 (matches reference)
  }
}

extern "C" void kernel_launch(void* const* d_in, const int* in_sizes, int n_in,
                              void* d_out, int out_size, void* d_ws, size_t ws_size,
                              hipStream_t stream) {
  const float* feat = (const float*)d_in[0];   // [4,256,38,38] f32
  const float* rois = (const float*)d_in[1];   // [64,5] f32
  float* out = (float*)d_out;                  // [64,256,7,7] f32

  const int N = in_sizes[1] / 5;               // 64 RoIs
  dim3 grid(N, C_ / CB_);                      // 64 x 32 blocks
  roipool_tdm_kernel<<<grid, 256, 0, stream>>>(feat, rois, out);
}